// Involution_24094766530811
// MI455X (gfx1250) — compile-verified
//
#include <hip/hip_runtime.h>

typedef __attribute__((ext_vector_type(2))) float v2f;
typedef __attribute__((ext_vector_type(8))) float v8f;

#define KSZ   7
#define PAD   3
#define GRP   16
#define CIN   256
#define CR    64        // CIN/4
#define KKG   784       // K*K*G
#define HH    56
#define WW    56
#define HW    (HH*WW)   // 3136
#define NB    4
#define PXB   32        // pixels per fused block
#define EPSV  1e-5f

// ---------------------------------------------------------------------------
// Kernel 1: T[n][o][p] = relu(BN(sum_c w1[o][c]*x[n][c][p] + b1[o]))
// One wave32 computes one 16(o) x 16(p) tile via V_WMMA_F32_16X16X4_F32.
// ---------------------------------------------------------------------------
__global__ __launch_bounds__(256) void involution_gemm1(
    const float* __restrict__ x,  const float* __restrict__ w1,
    const float* __restrict__ b1, const float* __restrict__ gamma,
    const float* __restrict__ beta, const float* __restrict__ mean,
    const float* __restrict__ var, float* __restrict__ T)
{
    const int lane = threadIdx.x & 31;
    const int wave = threadIdx.x >> 5;
    int tile = blockIdx.x * 8 + wave;        // [0, 3136)
    const int ptile = tile % 196; tile /= 196;
    const int otile = tile % 4;
    const int n     = tile / 4;
    const int p0 = ptile * 16, o0 = otile * 16;
    const int half = lane >> 4;              // 0: K={0,1}, 1: K={2,3}
    const int mn   = lane & 15;              // M row (for A) / N col (for B)

    const float* xr = x  + (size_t)n * CIN * HW + p0 + mn;
    const float* ar = w1 + (size_t)(o0 + mn) * CIN;

    v8f acc = {};
    for (int k = 0; k < CIN; k += 4) {
        v2f a, b;
        a.x = ar[k + 2 * half];
        a.y = ar[k + 2 * half + 1];
        b.x = xr[(size_t)(k + 2 * half) * HW];
        b.y = xr[(size_t)(k + 2 * half + 1) * HW];
        acc = __builtin_amdgcn_wmma_f32_16x16x4_f32(
                  false, a, false, b, (short)0, acc, false, false);
    }

    float* Tout = T + (size_t)n * CR * HW;
#pragma unroll
    for (int v = 0; v < 8; ++v) {
        const int o = o0 + v + 8 * half;
        const float scale = rsqrtf(var[o] + EPSV) * gamma[o];
        float val = (acc[v] + b1[o] - mean[o]) * scale + beta[o];
        val = fmaxf(val, 0.0f);
        Tout[(size_t)o * HW + p0 + mn] = val;
    }
}

// ---------------------------------------------------------------------------
// Kernel 2 (fused GEMM2 + involution apply), one block per (n, 32-pixel strip):
//   1. async-stage T tile (64 x 32 = 8 KB) into LDS via
//      GLOBAL_LOAD_ASYNC_TO_LDS_B128 (ASYNCcnt) + s_wait_asynccnt + barrier
//   2. 8 waves cover 49 o-tiles of  Ker(784x32) = w2(784x64) x Ttile(64x32);
//      each w2 A-fragment load feeds TWO v_wmma_f32_16x16x4_f32 (two
//      N-subtiles, two accumulators); 100 KB Ker slab stays in LDS
//   3. 256 threads apply the 49-tap involution for 256 channels x 32 px
// ---------------------------------------------------------------------------
__global__ __launch_bounds__(256) void involution_gemm2_apply(
    const float* __restrict__ T, const float* __restrict__ w2,
    const float* __restrict__ b2, const float* __restrict__ x,
    float* __restrict__ out)
{
    __shared__ float Tt[CR * PXB];           // 8 KB    : T tile   [k][px]
    __shared__ float kbuf[KKG * PXB];        // 100 KB  : Ker tile [o][px]

    const int ptile = blockIdx.x % (HW / PXB);   // 98 strips
    const int n     = blockIdx.x / (HW / PXB);
    const int p0    = ptile * PXB;

    // ---- stage T tile: 256 lanes x 2 x 16B = 8 KB, all 16B-aligned ----
#pragma unroll
    for (int j = 0; j < 2; ++j) {
        const int idx = (threadIdx.x + j * 256) * 4;   // float index in tile
        const int k = idx >> 5, px = idx & (PXB - 1);
        const float* gp = T + ((size_t)n * CR + k) * HW + p0 + px;
        unsigned ldsoff = (unsigned)(uintptr_t)&Tt[idx];
        asm volatile("global_load_async_to_lds_b128 %0, %1, off"
                     :: "v"(ldsoff), "v"(gp) : "memory");
    }
    asm volatile("s_wait_asynccnt 0" ::: "memory");
    __syncthreads();

    // ---- GEMM: Ker tile = w2 x Tt; A reused across two N-subtiles ----
    const int lane = threadIdx.x & 31;
    const int wave = threadIdx.x >> 5;
    const int half = lane >> 4;
    const int mn   = lane & 15;

    for (int ot = wave; ot < 49; ot += 8) {
        const int o0 = ot * 16;
        const float* ar = w2 + (size_t)(o0 + mn) * CR;
        v8f acc0 = {}, acc1 = {};
#pragma unroll
        for (int k = 0; k < CR; k += 4) {
            v2f a, b0, b1v;
            a.x  = ar[k + 2 * half];
            a.y  = ar[k + 2 * half + 1];
            b0.x = Tt[(k + 2 * half) * PXB + mn];
            b0.y = Tt[(k + 2 * half + 1) * PXB + mn];
            b1v.x = Tt[(k + 2 * half) * PXB + 16 + mn];
            b1v.y = Tt[(k + 2 * half + 1) * PXB + 16 + mn];
            acc0 = __builtin_amdgcn_wmma_f32_16x16x4_f32(
                       false, a, false, b0, (short)0, acc0, false, false);
            acc1 = __builtin_amdgcn_wmma_f32_16x16x4_f32(
                       false, a, false, b1v, (short)0, acc1, false, false);
        }
#pragma unroll
        for (int v = 0; v < 8; ++v) {
            const int o = o0 + v + 8 * half;
            const float bias = b2[o];
            kbuf[o * PXB + mn]      = acc0[v] + bias;
            kbuf[o * PXB + 16 + mn] = acc1[v] + bias;
        }
    }
    __syncthreads();

    // ---- involution apply: 32 px x 256 channels from this block ----
    const int px = threadIdx.x & (PXB - 1);   // [0,32)
    const int cg = threadIdx.x >> 5;          // [0,8): channel sub-group
    const int p  = p0 + px;
    const int h  = p / WW;
    const int w  = p % WW;

    for (int g = 0; g < GRP; ++g) {           // kernel group
#pragma unroll
        for (int s = 0; s < 2; ++s) {
            const int c = g * GRP + cg + 8 * s;
            const float* xc = x + ((size_t)n * CIN + c) * HW;
            const float* kp = &kbuf[(g * 49) * PXB + px];
            float acc = 0.0f;
#pragma unroll
            for (int kh = 0; kh < KSZ; ++kh) {
                const int hy = h + kh - PAD;
                if (hy < 0 || hy >= HH) continue;
#pragma unroll
                for (int kw = 0; kw < KSZ; ++kw) {
                    const int wx = w + kw - PAD;
                    if (wx < 0 || wx >= WW) continue;
                    acc = fmaf(xc[(size_t)hy * WW + wx],
                               kp[(kh * KSZ + kw) * PXB], acc);
                }
            }
            out[((size_t)n * CIN + c) * HW + p] = acc;
        }
    }
}

// ---------------------------------------------------------------------------
extern "C" void kernel_launch(void* const* d_in, const int* in_sizes, int n_in,
                              void* d_out, int out_size, void* d_ws, size_t ws_size,
                              hipStream_t stream)
{
    const float* x     = (const float*)d_in[0];
    const float* w1    = (const float*)d_in[1];
    const float* b1    = (const float*)d_in[2];
    const float* gamma = (const float*)d_in[3];
    const float* beta  = (const float*)d_in[4];
    const float* mean  = (const float*)d_in[5];
    const float* var   = (const float*)d_in[6];
    const float* w2    = (const float*)d_in[7];
    const float* b2    = (const float*)d_in[8];

    float* T = (float*)d_ws;                 // 4*64*3136 floats = 3.2 MB

    involution_gemm1<<<392, 256, 0, stream>>>(x, w1, b1, gamma, beta, mean, var, T);
    involution_gemm2_apply<<<NB * (HW / PXB), 256, 0, stream>>>(
        T, w2, b2, x, (float*)d_out);
}